// InterpretableMultiHeadAttention_5549097746841
// MI455X (gfx1250) — compile-verified
//
#include <hip/hip_runtime.h>

// MI455X / gfx1250: wave32, WMMA (f32 16x16x4 keeps full precision; compute is
// negligible here — the kernel is bound by the 537MB attn_out store).
// attn_out / out stores are non-temporal (TH=NT) so the streaming 537MB write
// does not evict the L2-resident qs/ks/vs working set (38MB, reused S/16 times).

typedef float v2f __attribute__((ext_vector_type(2)));
typedef float v4f __attribute__((ext_vector_type(4)));
typedef float v8f __attribute__((ext_vector_type(8)));

#define S_LEN 2048
#define NH    4
#define DK    64
#define DM    256
#define BATCH 8

// D = A(16x4,f32) x B(4x16,f32) + C(16x16,f32)
// A layout: lanes0-15 row=lane K={0,1}; lanes16-31 row=lane-16 K={2,3}
// B layout: lanes0-15 col=lane K={0,1}; lanes16-31 col=lane-16 K={2,3}
// C/D     : lane%16=col; vgpr r = row r (lanes<16) / row r+8 (lanes>=16)
__device__ __forceinline__ v8f wmma_f32(v2f a, v2f b, v8f c) {
  return __builtin_amdgcn_wmma_f32_16x16x4_f32(false, a, false, b, (short)0, c,
                                               false, false);
}

// ---------------------------------------------------------------------------
// Kernel 1: projections. grid.x = (B*S)/16 row tiles, grid.y = 9:
//   0..3 -> qs[h] (prescaled by 1/sqrt(64)), 4..7 -> ks[h], 8 -> vs.
// 4 waves per block, each wave owns a 16-wide slice of the 64 output cols.
// ---------------------------------------------------------------------------
__global__ void __launch_bounds__(128) proj_kernel(
    const float* __restrict__ q, const float* __restrict__ k,
    const float* __restrict__ v, const float* __restrict__ Wq,
    const float* __restrict__ bq, const float* __restrict__ Wk,
    const float* __restrict__ bk, const float* __restrict__ Wv,
    const float* __restrict__ bv, float* __restrict__ qs,
    float* __restrict__ ks, float* __restrict__ vs) {
  const int lane = threadIdx.x & 31;
  const int wave = threadIdx.x >> 5;
  const int t = blockIdx.x;  // row tile over flattened B*S
  const int p = blockIdx.y;

  const float* x;
  const float* W;
  const float* bias;
  int head;
  float scale;
  if (p < 4) {
    x = q; W = Wq + (size_t)p * DK * DM; bias = bq + p * DK; head = p;
    scale = 0.125f;  // fold 1/sqrt(d_k) into qs
  } else if (p < 8) {
    x = k; W = Wk + (size_t)(p - 4) * DK * DM; bias = bk + (p - 4) * DK;
    head = p - 4; scale = 1.0f;
  } else {
    x = v; W = Wv; bias = bv; head = 0; scale = 1.0f;
  }

  const int e0 = wave * 16;
  const int klo = (lane < 16) ? 0 : 2;
  const int r16 = lane & 15;

  v8f acc = {};
  const float* arow = x + (size_t)(t * 16 + r16) * DM + klo;
  const float* brow = W + (size_t)(e0 + r16) * DM + klo;  // B[kd][e] = W[e][kd]
#pragma unroll
  for (int d0 = 0; d0 < DM; d0 += 4) {
    v2f a = *(const v2f*)(arow + d0);
    v2f b = *(const v2f*)(brow + d0);
    acc = wmma_f32(a, b, acc);
  }

  const int col = e0 + r16;
  const float bval = bias[col];
#pragma unroll
  for (int i = 0; i < 8; ++i) acc[i] = (acc[i] + bval) * scale;

#pragma unroll
  for (int i = 0; i < 8; ++i) {
    const int m = i + ((lane < 16) ? 0 : 8);
    const int gr = t * 16 + m;
    if (p < 8) {
      const int b_ = gr >> 11, s_ = gr & 2047;
      const size_t idx = ((size_t)(b_ * NH + head) * S_LEN + s_) * DK + col;
      ((p < 4) ? qs : ks)[idx] = acc[i];
    } else {
      vs[(size_t)gr * DK + col] = acc[i];
    }
  }
}

// ---------------------------------------------------------------------------
// Kernel 2: fused attention per (b, h, 16-query tile). 4 waves split the 128
// key tiles. Two-pass online softmax (recompute QK^T instead of re-reading
// 537MB of attn). Score tile computed TRANSPOSED (D[k][q] = K x Q^T) so each
// lane owns one query row -> cheap softmax + contiguous b128 attn stores.
// ---------------------------------------------------------------------------
__global__ void __launch_bounds__(128) attn_kernel(
    const float* __restrict__ qs, const float* __restrict__ ks,
    const float* __restrict__ vs, float* __restrict__ attn,
    float* __restrict__ Oh) {
  const int lane = threadIdx.x & 31;
  const int wave = threadIdx.x >> 5;
  const int qt = blockIdx.x, h = blockIdx.y, b = blockIdx.z;
  const int q0 = qt * 16;
  const int klo = (lane < 16) ? 0 : 2;
  const int row = lane & 15;     // query row owned by this lane
  const int hf = (lane < 16) ? 0 : 1;

  const float* qbh = qs + (size_t)(b * NH + h) * S_LEN * DK;
  const float* kbh = ks + (size_t)(b * NH + h) * S_LEN * DK;
  const float* vb = vs + (size_t)b * S_LEN * DK;

  // Q tile as resident B-operand: qb[st] = {qs[q0+row][4*st+klo], +1}
  v2f qb[16];
  {
    const float* qrow = qbh + (size_t)(q0 + row) * DK + klo;
#pragma unroll
    for (int st = 0; st < 16; ++st) qb[st] = *(const v2f*)(qrow + 4 * st);
  }

  const int kt0 = wave * 32, kt1 = kt0 + 32;  // 128 key tiles / 4 waves

  __shared__ float red_m[4][16];
  __shared__ float red_l[4][16];
  __shared__ float Obuf[4][16 * DK];

  // ---- pass 1: row max + denominator (online) ----
  float m_run = -3.0e38f, l_run = 0.0f;
  for (int kt = kt0; kt < kt1; ++kt) {
    v8f acc = {};
    const float* krow = kbh + (size_t)(kt * 16 + row) * DK + klo;
#pragma unroll
    for (int st = 0; st < 16; ++st) {
      v2f a = *(const v2f*)(krow + 4 * st);
      acc = wmma_f32(a, qb[st], acc);  // D[k][q], scale already in qs
    }
    float mt = acc[0];
#pragma unroll
    for (int i = 1; i < 8; ++i) mt = fmaxf(mt, acc[i]);
    mt = fmaxf(mt, __shfl_xor(mt, 16, 32));  // combine k-halves
    const float mnew = fmaxf(m_run, mt);
    float ps = 0.0f;
#pragma unroll
    for (int i = 0; i < 8; ++i) ps += __expf(acc[i] - mnew);
    ps += __shfl_xor(ps, 16, 32);
    l_run = l_run * __expf(m_run - mnew) + ps;
    m_run = mnew;
  }

  if (lane < 16) { red_m[wave][lane] = m_run; red_l[wave][lane] = l_run; }
  __syncthreads();
  float M = red_m[0][row];
#pragma unroll
  for (int w = 1; w < 4; ++w) M = fmaxf(M, red_m[w][row]);
  float L = 0.0f;
#pragma unroll
  for (int w = 0; w < 4; ++w) L += red_l[w][row] * __expf(red_m[w][row] - M);
  const float invL = 1.0f / L;

  // ---- pass 2: normalized P -> global store + P x V accumulation ----
  v8f accO[4] = {};  // O[16 x 64], 4 column tiles
  for (int kt = kt0; kt < kt1; ++kt) {
    v8f acc = {};
    const float* krow = kbh + (size_t)(kt * 16 + row) * DK + klo;
#pragma unroll
    for (int st = 0; st < 16; ++st) {
      v2f a = *(const v2f*)(krow + 4 * st);
      acc = wmma_f32(a, qb[st], acc);
    }
    float p[8];
#pragma unroll
    for (int i = 0; i < 8; ++i) p[i] = __expf(acc[i] - M) * invL;

    // attn[b, q0+row, h, kt*16 + hf*8 + i] : 8 consecutive k -> 2x b128.
    // Non-temporal: 537MB streaming write must not flush K/V out of L2.
    {
      const size_t base =
          ((size_t)(b * S_LEN + q0 + row) * NH + h) * S_LEN + kt * 16 + hf * 8;
      v4f lo, hi;
      lo.x = p[0]; lo.y = p[1]; lo.z = p[2]; lo.w = p[3];
      hi.x = p[4]; hi.y = p[5]; hi.z = p[6]; hi.w = p[7];
      __builtin_nontemporal_store(lo, (v4f*)(attn + base));
      __builtin_nontemporal_store(hi, (v4f*)(attn + base + 4));
    }

    // O += P(16x16) x V(16x64): convert P (D-layout of transposed tile) into
    // A-layout via ds_bpermute shuffles, 4 K-substeps of 4.
#pragma unroll
    for (int kk = 0; kk < 4; ++kk) {
      const int srcl = row + ((kk < 2) ? 0 : 16);
      const float g0 = __shfl(p[(kk & 1) * 4 + 0], srcl, 32);
      const float g1 = __shfl(p[(kk & 1) * 4 + 1], srcl, 32);
      const float g2 = __shfl(p[(kk & 1) * 4 + 2], srcl, 32);
      const float g3 = __shfl(p[(kk & 1) * 4 + 3], srcl, 32);
      v2f a;
      a.x = (lane < 16) ? g0 : g2;
      a.y = (lane < 16) ? g1 : g3;
      const float* vrow = vb + (size_t)(kt * 16 + kk * 4 + klo) * DK + row;
#pragma unroll
      for (int et = 0; et < 4; ++et) {
        v2f bb;
        bb.x = vrow[et * 16];
        bb.y = vrow[et * 16 + DK];
        accO[et] = wmma_f32(a, bb, accO[et]);
      }
    }
  }

  // reduce partial O across the 4 waves via LDS, store per-head O
#pragma unroll
  for (int et = 0; et < 4; ++et)
#pragma unroll
    for (int i = 0; i < 8; ++i) {
      const int qq = i + ((lane < 16) ? 0 : 8);
      Obuf[wave][qq * DK + et * 16 + row] = accO[et][i];
    }
  __syncthreads();
  {
    float* dst = Oh + ((size_t)(b * NH + h) * S_LEN + q0) * DK;
#pragma unroll
    for (int j = 0; j < 8; ++j) {
      const int idx = threadIdx.x * 8 + j;
      dst[idx] = Obuf[0][idx] + Obuf[1][idx] + Obuf[2][idx] + Obuf[3][idx];
    }
  }
}

// ---------------------------------------------------------------------------
// Kernel 3: out[b,s,:] = (mean_h O[b,h,s,:]) @ Wh^T. grid = (B*S)/16 tiles,
// 4 waves x 4 column tiles of 16 cover d_model=256. K = 64 -> 16 WMMA steps.
// ---------------------------------------------------------------------------
__global__ void __launch_bounds__(128) out_kernel(const float* __restrict__ Oh,
                                                  const float* __restrict__ Wh,
                                                  float* __restrict__ out) {
  const int lane = threadIdx.x & 31;
  const int wave = threadIdx.x >> 5;
  const int t = blockIdx.x;
  const int klo = (lane < 16) ? 0 : 2;
  const int r16 = lane & 15;

  const int gr = t * 16 + r16;
  const int b_ = gr >> 11, s_ = gr & 2047;

  v2f amat[16];
#pragma unroll
  for (int st = 0; st < 16; ++st) {
    float sx = 0.0f, sy = 0.0f;
#pragma unroll
    for (int hh = 0; hh < 4; ++hh) {
      const v2f x = *(const v2f*)(Oh + ((size_t)(b_ * NH + hh) * S_LEN + s_) * DK
                                  + 4 * st + klo);
      sx += x.x;
      sy += x.y;
    }
    amat[st].x = sx * 0.25f;
    amat[st].y = sy * 0.25f;
  }

  for (int nt = wave * 4; nt < wave * 4 + 4; ++nt) {
    const int d0 = nt * 16;
    v8f acc = {};
    // B[e][d] = Wh[d][e]; Wh is [256,64] row-major -> b64 along e
    const float* wrow = Wh + (size_t)(d0 + r16) * DK + klo;
#pragma unroll
    for (int st = 0; st < 16; ++st) {
      const v2f bb = *(const v2f*)(wrow + 4 * st);
      acc = wmma_f32(amat[st], bb, acc);
    }
    const int col = d0 + r16;
#pragma unroll
    for (int i = 0; i < 8; ++i) {
      const int m = i + ((lane < 16) ? 0 : 8);
      // write-once output: non-temporal store
      __builtin_nontemporal_store(acc[i],
                                  out + (size_t)(t * 16 + m) * DM + col);
    }
  }
}

// ---------------------------------------------------------------------------
extern "C" void kernel_launch(void* const* d_in, const int* in_sizes, int n_in,
                              void* d_out, int out_size, void* d_ws,
                              size_t ws_size, hipStream_t stream) {
  const float* q = (const float*)d_in[0];
  const float* k = (const float*)d_in[1];
  const float* v = (const float*)d_in[2];
  const float* Wq = (const float*)d_in[3];
  const float* bq = (const float*)d_in[4];
  const float* Wk = (const float*)d_in[5];
  const float* bk = (const float*)d_in[6];
  const float* Wv = (const float*)d_in[7];
  const float* bv = (const float*)d_in[8];
  const float* Wh = (const float*)d_in[9];

  float* out = (float*)d_out;                         // [B,S,256]
  float* attn = out + (size_t)BATCH * S_LEN * DM;     // [B,Sq,H,Sk]

  // workspace: qs(16.8MB) ks(16.8MB) vs(4.2MB) Oh(16.8MB) ~= 54.6MB
  float* qs = (float*)d_ws;
  float* ks = qs + (size_t)BATCH * NH * S_LEN * DK;
  float* vs = ks + (size_t)BATCH * NH * S_LEN * DK;
  float* Oh = vs + (size_t)BATCH * S_LEN * DK;

  proj_kernel<<<dim3(BATCH * S_LEN / 16, 9), 128, 0, stream>>>(
      q, k, v, Wq, bq, Wk, bk, Wv, bv, qs, ks, vs);
  attn_kernel<<<dim3(S_LEN / 16, NH, BATCH), 128, 0, stream>>>(qs, ks, vs,
                                                               attn, Oh);
  out_kernel<<<dim3(BATCH * S_LEN / 16), 128, 0, stream>>>(Oh, Wh, out);
}